// KnnFeature_57251914056092
// MI455X (gfx1250) — compile-verified
//
#include <hip/hip_runtime.h>

// Problem constants (match reference)
#define BATCH 4
#define CIN   3
#define NPTS  8192
#define KNN   16
#define DOUT  64
#define EPSBN 1e-5f
#define SLOPE 0.2f
#define CH    2048   // LDS column-chunk size for knn

typedef __attribute__((ext_vector_type(2))) float v2f;
typedef __attribute__((ext_vector_type(8))) float v8f;

__device__ __forceinline__ v8f wmma4(v2f a, v2f b, v8f c) {
    // D = A(16x4 f32) * B(4x16 f32) + C(16x16 f32)
    return __builtin_amdgcn_wmma_f32_16x16x4_f32(
        /*neg_a=*/false, a, /*neg_b=*/false, b,
        /*c_mod=*/(short)0, c, /*reuse_a=*/false, /*reuse_b=*/false);
}

// ---------------------------------------------------------------------------
// Kernel 1: KNN via WMMA Gram tiles. One wave owns 32 query rows (2 WMMAs per
// 16-column tile) so the top-k scan keeps all 32 lanes busy (one row each).
// grid = BATCH*NPTS/256 blocks of 256 threads (8 waves -> 256 rows/block).
// ---------------------------------------------------------------------------
__global__ __launch_bounds__(256) void knn_kernel(const float* __restrict__ feats,
                                                  int* __restrict__ idx_out,
                                                  float* __restrict__ dist_out) {
    __shared__ __align__(16) float sx[CH], sy[CH], sz[CH], sq[CH];  // 32 KB
    __shared__ __align__(16) float tile_g[8][32][20];               // 20 KB, rows 16B-aligned

    const int wave = threadIdx.x >> 5;
    const int lane = threadIdx.x & 31;
    const int b = blockIdx.x >> 5;                    // 32 blocks per batch
    const int rowBase = ((blockIdx.x & 31) << 8) + (wave << 5);   // 32 rows per wave
    const float* fb = feats + (size_t)b * CIN * NPTS;

    const int r = lane & 15;
    const bool hi = lane >= 16;

    // A fragment rows: frag0 = rowBase+0..15, frag1 = rowBase+16..31
    {
    }
    const int fr0 = rowBase + r;
    const int fr1 = rowBase + 16 + r;
    float f0x = fb[fr0], f0y = fb[NPTS + fr0], f0z = fb[2 * NPTS + fr0];
    float f1x = fb[fr1], f1y = fb[NPTS + fr1], f1z = fb[2 * NPTS + fr1];
    v2f af0, af1;
    af0.x = hi ? f0z : f0x;  af0.y = hi ? 0.0f : f0y;
    af1.x = hi ? f1z : f1x;  af1.y = hi ? 0.0f : f1y;

    // Each lane owns row rowBase + lane for the top-k scan
    const int myRow = rowBase + lane;
    const float rx = fb[myRow], ry = fb[NPTS + myRow], rz = fb[2 * NPTS + myRow];
    const float rowSq = rx * rx + ry * ry + rz * rz;

    float bd[16];
    int   bi[16];
#pragma unroll
    for (int s = 0; s < 16; ++s) { bd[s] = 3.4e38f; bi[s] = 0; }
    float worst = 3.4e38f;

    for (int cbase = 0; cbase < NPTS; cbase += CH) {
        __syncthreads();
        for (int i = threadIdx.x; i < CH; i += 256) {
            float px = fb[cbase + i];
            float py = fb[NPTS + cbase + i];
            float pz = fb[2 * NPTS + cbase + i];
            sx[i] = px; sy[i] = py; sz[i] = pz;
            sq[i] = px * px + py * py + pz * pz;
            // prefetch next chunk while this one streams in
            if (cbase + CH < NPTS)
                __builtin_prefetch(&fb[cbase + CH + i], 0, 1);
        }
        __syncthreads();

        for (int m0 = 0; m0 < CH; m0 += 16) {
            const int cl = m0 + r;
            // load all coords unconditionally, select with cndmask (no exec juggling)
            const float bx = sx[cl], by = sy[cl], bz = sz[cl];
            v2f bf;
            bf.x = hi ? bz : bx;
            bf.y = hi ? 0.0f : by;

            v8f g0 = {}, g1 = {};
            g0 = wmma4(af0, bf, g0);   // rows 0..15  x cols m0..m0+15
            g1 = wmma4(af1, bf, g1);   // rows 16..31 x cols m0..m0+15

            // D layout: vgpr i -> row i (lanes 0-15) / row i+8 (lanes 16-31), col = r
#pragma unroll
            for (int i = 0; i < 8; ++i) {
                tile_g[wave][i + (hi ? 8 : 0)][r] = g0[i];
                tile_g[wave][16 + i + (hi ? 8 : 0)][r] = g1[i];
            }
            asm volatile("s_wait_dscnt 0" ::: "memory");  // wave-local LDS RAW

            // scan: every lane owns one row -> fully converged
            const float4* grow = (const float4*)&tile_g[wave][lane][0];
            const float4* srow = (const float4*)&sq[m0];
#pragma unroll
            for (int q = 0; q < 4; ++q) {
                float4 g4 = grow[q];
                float4 s4 = srow[q];   // lane-uniform -> LDS broadcast
#pragma unroll
                for (int t = 0; t < 4; ++t) {
                    float g  = (t == 0) ? g4.x : (t == 1) ? g4.y : (t == 2) ? g4.z : g4.w;
                    float sv = (t == 0) ? s4.x : (t == 1) ? s4.y : (t == 2) ? s4.z : s4.w;
                    float cd = fmaxf(rowSq + sv - 2.0f * g, 0.0f);
                    int ci = cbase + m0 + q * 4 + t;
                    if (cd < worst) {
                        int slot = 0; float mx = bd[0];
#pragma unroll
                        for (int s = 1; s < 16; ++s)
                            if (bd[s] > mx) { mx = bd[s]; slot = s; }
#pragma unroll
                        for (int s = 0; s < 16; ++s)
                            if (s == slot) { bd[s] = cd; bi[s] = ci; }
                        mx = bd[0];
#pragma unroll
                        for (int s = 1; s < 16; ++s) mx = fmaxf(mx, bd[s]);
                        worst = mx;
                    }
                }
            }
        }
    }

    // ascending sort by (dist, idx) — matches top_k ordering; all lanes active
#pragma unroll
    for (int i = 0; i < 15; ++i) {
#pragma unroll
        for (int j = i + 1; j < 16; ++j) {
            bool sw = (bd[j] < bd[i]) || (bd[j] == bd[i] && bi[j] < bi[i]);
            float td = sw ? bd[j] : bd[i];
            float ud = sw ? bd[i] : bd[j];
            int   ti = sw ? bi[j] : bi[i];
            int   ui = sw ? bi[i] : bi[j];
            bd[i] = td; bd[j] = ud; bi[i] = ti; bi[j] = ui;
        }
    }
    const size_t ob = ((size_t)(b * NPTS + myRow)) * KNN;
#pragma unroll
    for (int s = 0; s < 16; ++s) {
        idx_out[ob + s] = bi[s];
        dist_out[ob + s] = bd[s];
    }
}

// ---------------------------------------------------------------------------
// Kernel 2: edge-feature + 1x1 conv via WMMA. One wave per (b,n): rows = k=0..15,
// cols = 64 output channels (4 tiles). 12 WMMAs per wave.
// grid = (BATCH*NPTS)/8 blocks of 256 threads.
// ---------------------------------------------------------------------------
__global__ __launch_bounds__(256) void conv_kernel(const float* __restrict__ feats,
                                                   const int* __restrict__ knn_idx,
                                                   const float* __restrict__ knn_dist,
                                                   const float* __restrict__ Wg,
                                                   float* __restrict__ out) {
    __shared__ float sW[DOUT * 12];   // W[o][c], 3 KB
    for (int i = threadIdx.x; i < DOUT * 12; i += 256) sW[i] = Wg[i];
    __syncthreads();

    const int wave = threadIdx.x >> 5;
    const int lane = threadIdx.x & 31;
    const int tile = blockIdx.x * 8 + wave;      // flat (b,n)
    const int b = tile >> 13;
    const int n = tile & (NPTS - 1);
    const int k = lane & 15;
    const bool hi = lane >= 16;

    const float* fb = feats + (size_t)b * CIN * NPTS;
    const float x0 = fb[n], x1 = fb[NPTS + n], x2 = fb[2 * NPTS + n];

    const size_t kb = ((size_t)(b * NPTS + n)) * KNN + k;
    const int ni = knn_idx[kb];
    const float d = knn_dist[kb];
    const float n0 = fb[ni], n1 = fb[NPTS + ni], n2 = fb[2 * NPTS + ni];

    float f[12] = { x0, x1, x2, n0, n1, n2, n0 - x0, n1 - x1, n2 - x2, d, d, d };

    v8f acc[4] = {};
    const int col = lane & 15;
#pragma unroll
    for (int cb = 0; cb < 12; cb += 4) {
        v2f a;
        a.x = hi ? f[cb + 2] : f[cb + 0];
        a.y = hi ? f[cb + 3] : f[cb + 1];
#pragma unroll
        for (int ot = 0; ot < 4; ++ot) {
            const int o = ot * 16 + col;
            // unconditional loads + cndmask selects
            const float w0 = sW[o * 12 + cb + 0];
            const float w1 = sW[o * 12 + cb + 1];
            const float w2 = sW[o * 12 + cb + 2];
            const float w3 = sW[o * 12 + cb + 3];
            v2f bw;
            bw.x = hi ? w2 : w0;
            bw.y = hi ? w3 : w1;
            acc[ot] = wmma4(a, bw, acc[ot]);
        }
    }

#pragma unroll
    for (int ot = 0; ot < 4; ++ot) {
        const int o = ot * 16 + col;
        float* op = out + (((size_t)(b * DOUT + o)) * NPTS + n) * KNN;
#pragma unroll
        for (int i = 0; i < 8; ++i)
            op[i + (hi ? 8 : 0)] = acc[ot][i];
    }
}

// ---------------------------------------------------------------------------
// Kernel 3: zero the BN stats accumulators (must run every launch).
// ---------------------------------------------------------------------------
__global__ void init_sums(float* sums) {
    if (threadIdx.x < 128) sums[threadIdx.x] = 0.0f;
}

// ---------------------------------------------------------------------------
// Kernel 4: per-channel sum / sumsq. One block per (b,o) slice.
// ---------------------------------------------------------------------------
__global__ __launch_bounds__(256) void reduce_kernel(const float* __restrict__ y,
                                                     float* __restrict__ sums) {
    const int o = blockIdx.x & (DOUT - 1);
    const int b = blockIdx.x >> 6;
    const float4* p4 = (const float4*)(y + ((size_t)(b * DOUT + o)) * (NPTS * KNN));
    float s = 0.0f, s2 = 0.0f;
    for (int i = threadIdx.x; i < (NPTS * KNN) / 4; i += 256) {
        float4 v = p4[i];
        s  += v.x + v.y + v.z + v.w;
        s2 += v.x * v.x + v.y * v.y + v.z * v.z + v.w * v.w;
    }
    __shared__ float ls[256], ls2[256];
    ls[threadIdx.x] = s; ls2[threadIdx.x] = s2;
    __syncthreads();
    for (int st = 128; st > 0; st >>= 1) {
        if (threadIdx.x < st) {
            ls[threadIdx.x]  += ls[threadIdx.x + st];
            ls2[threadIdx.x] += ls2[threadIdx.x + st];
        }
        __syncthreads();
    }
    if (threadIdx.x == 0) {
        atomicAdd(&sums[o], ls[0]);
        atomicAdd(&sums[DOUT + o], ls2[0]);
    }
}

// ---------------------------------------------------------------------------
// Kernel 5: BN (batch stats, biased var) + affine + LeakyReLU, in place.
// ---------------------------------------------------------------------------
__global__ __launch_bounds__(256) void finalize_kernel(float* __restrict__ y,
                                                       const float* __restrict__ sums,
                                                       const float* __restrict__ gamma,
                                                       const float* __restrict__ beta) {
    const size_t i = ((size_t)blockIdx.x * 256 + threadIdx.x) * 4;
    const int o = (int)((i >> 17) & (DOUT - 1));      // NPTS*KNN = 2^17
    const float inv = 1.0f / (float)((size_t)BATCH * NPTS * KNN);
    const float mean = sums[o] * inv;
    const float var = sums[DOUT + o] * inv - mean * mean;
    const float sc = gamma[o] * rsqrtf(var + EPSBN);
    const float sh = beta[o] - mean * sc;
    float4 v = *(float4*)(y + i);
    float t0 = v.x * sc + sh, t1 = v.y * sc + sh, t2 = v.z * sc + sh, t3 = v.w * sc + sh;
    v.x = t0 >= 0.0f ? t0 : SLOPE * t0;
    v.y = t1 >= 0.0f ? t1 : SLOPE * t1;
    v.z = t2 >= 0.0f ? t2 : SLOPE * t2;
    v.w = t3 >= 0.0f ? t3 : SLOPE * t3;
    *(float4*)(y + i) = v;
}

extern "C" void kernel_launch(void* const* d_in, const int* in_sizes, int n_in,
                              void* d_out, int out_size, void* d_ws, size_t ws_size,
                              hipStream_t stream) {
    const float* feats = (const float*)d_in[0];   // [B, C, N, 1]
    const float* W     = (const float*)d_in[1];   // [DOUT, 12]
    const float* gamma = (const float*)d_in[2];
    const float* beta  = (const float*)d_in[3];
    float* out = (float*)d_out;                   // [B, DOUT, N, K]

    const size_t nIdx = (size_t)BATCH * NPTS * KNN;
    int*   knn_idx  = (int*)d_ws;
    float* knn_dist = (float*)((char*)d_ws + nIdx * sizeof(int));
    float* sums     = (float*)((char*)d_ws + nIdx * (sizeof(int) + sizeof(float)));

    knn_kernel<<<BATCH * (NPTS / 256), 256, 0, stream>>>(feats, knn_idx, knn_dist);
    conv_kernel<<<(BATCH * NPTS) / 8, 256, 0, stream>>>(feats, knn_idx, knn_dist, W, out);
    init_sums<<<1, 128, 0, stream>>>(sums);
    reduce_kernel<<<BATCH * DOUT, 256, 0, stream>>>(out, sums);
    finalize_kernel<<<(int)(((size_t)BATCH * DOUT * NPTS * KNN) / 1024), 256, 0, stream>>>(
        out, sums, gamma, beta);
}